// Decoder_13640816132672
// MI455X (gfx1250) — compile-verified
//
#include <hip/hip_runtime.h>
#include <math.h>
#include <stdint.h>

typedef __attribute__((ext_vector_type(2))) float v2f;
typedef __attribute__((ext_vector_type(8))) float v8f;

#define S_LEN 2048
#define NB    32
#define HD    512
#define EMB   256
#define G4    2048   // 4*H
#define NPART 8      // s-partitions per batch row
#define PROWS (S_LEN / NPART)   // 256 rows per partition
#define CHROWS 8                // rows staged in LDS per chunk

__device__ __forceinline__ float warp_red_sum(float v) {
  for (int off = 16; off > 0; off >>= 1) v += __shfl_down(v, off, 32);
  return v;
}

// hc[n] = dot(hidden[1][n], W_energy[0:H])
__global__ void k_hconst(const float* __restrict__ hidden, const float* __restrict__ Wen,
                         float* __restrict__ hc) {
  int n = blockIdx.x, lane = threadIdx.x;
  const float* hrow = hidden + NB * HD + n * HD;  // hidden[-1]
  float acc = 0.f;
  for (int j = lane; j < HD; j += 32) acc += hrow[j] * Wen[j];
  acc = warp_red_sum(acc);
  if (lane == 0) hc[n] = acc;
}

// Fused attention: single HBM pass over encoder_states, async-copied straight
// into LDS (GLOBAL_LOAD_ASYNC_TO_LDS_B128, ASYNCcnt) with double buffering so
// the next 32KB tile streams in while the current one is reduced.
// Block = (n, partition of 256 s-rows); online softmax over the partition.
__global__ void k_attn_fused(const float* __restrict__ enc, const float* __restrict__ Wen,
                             const float* __restrict__ hc, const float* __restrict__ be,
                             float* __restrict__ pC, float* __restrict__ pstats) {
  __shared__ __align__(16) float tile[2][CHROWS * 1024];  // 2 x 32 KB
  __shared__ float w2l[1024];                             // W_energy[H:3H]
  __shared__ float ev[CHROWS];

  int t = threadIdx.x;                    // 0..255
  int n = blockIdx.x >> 3;
  int part = blockIdx.x & (NPART - 1);
  int s0 = part * PROWS;
  int wv = t >> 5, ln = t & 31;

  for (int j = t; j < 1024; j += 256) w2l[j] = Wen[HD + j];
  float hcn = hc[n], bias = be[0];

  uint32_t ldsb[2];
  ldsb[0] = (uint32_t)(uintptr_t)(void*)&tile[0][0];  // low 32 bits = LDS offset
  ldsb[1] = (uint32_t)(uintptr_t)(void*)&tile[1][0];

  const int NCH = PROWS / CHROWS;  // 32 chunks

  auto issue = [&](int buf, int chunk) {
    int sbase = s0 + chunk * CHROWS;
#pragma unroll
    for (int i = t; i < CHROWS * 256; i += 256) {   // 8 B128 asyncs per wave
      int r = i >> 8, c4 = i & 255;
      uint64_t ga = (uint64_t)(uintptr_t)(enc + ((long)(sbase + r) * NB + n) * 1024 + c4 * 4);
      uint32_t la = ldsb[buf] + (uint32_t)((r * 1024 + c4 * 4) * 4);
      asm volatile("global_load_async_to_lds_b128 %0, %1, off"
                   :: "v"(la), "v"(ga) : "memory");
    }
  };

  issue(0, 0);

  float M = -1e30f, Z = 0.f;
  float4 C = make_float4(0.f, 0.f, 0.f, 0.f);

  for (int c = 0; c < NCH; ++c) {
    int cur = c & 1;
    if (c + 1 < NCH) {
      issue(cur ^ 1, c + 1);                          // prefetch next tile
      asm volatile("s_wait_asynccnt 0x8" ::: "memory");  // wait chunk c (in-order)
    } else {
      asm volatile("s_wait_asynccnt 0x0" ::: "memory");
    }
    __syncthreads();  // chunk c visible to all waves

    const float* tl = tile[cur];
    // wave wv computes energy of row wv from LDS
    float d = 0.f;
    for (int j = ln; j < 1024; j += 32) d += tl[wv * 1024 + j] * w2l[j];
    d = warp_red_sum(d);
    if (ln == 0) {
      float val = hcn + d + bias;
      ev[wv] = val > 0.f ? val : 0.f;   // ReLU => energies >= 0
    }
    __syncthreads();

    // online softmax update (block-uniform)
    float cm = ev[0];
#pragma unroll
    for (int r = 1; r < CHROWS; ++r) cm = fmaxf(cm, ev[r]);
    float Mnew = fmaxf(M, cm);
    float scale = expf(M - Mnew);       // first chunk: exp(-1e30) == 0
    Z *= scale;
    C.x *= scale; C.y *= scale; C.z *= scale; C.w *= scale;

    float pr[CHROWS];
#pragma unroll
    for (int r = 0; r < CHROWS; ++r) { pr[r] = expf(ev[r] - Mnew); Z += pr[r]; }
    M = Mnew;

    // accumulate 4 private context columns per thread from LDS
#pragma unroll
    for (int r = 0; r < CHROWS; ++r) {
      float4 v = *(const float4*)(tl + r * 1024 + t * 4);
      C.x += pr[r] * v.x; C.y += pr[r] * v.y;
      C.z += pr[r] * v.z; C.w += pr[r] * v.w;
    }
    __syncthreads();  // all reads of buffer `cur` done before it is re-issued
  }

  float* outC = pC + ((long)blockIdx.x) * 1024 + t * 4;
  *(float4*)outC = C;
  if (t == 0) { pstats[2 * blockIdx.x] = M; pstats[2 * blockIdx.x + 1] = Z; }
}

// Merge NPART partials per n: ctx[n][c] = sum_p C_p[c]*exp(M_p-Mg) / Zg
__global__ void k_attn_combine(const float* __restrict__ pC, const float* __restrict__ pstats,
                               float* __restrict__ ctx) {
  int n = blockIdx.x, t = threadIdx.x;
  float Mg = -1e30f;
#pragma unroll
  for (int p = 0; p < NPART; ++p) Mg = fmaxf(Mg, pstats[2 * (n * NPART + p)]);
  float Zg = 0.f;
  float w[NPART];
#pragma unroll
  for (int p = 0; p < NPART; ++p) {
    w[p] = expf(pstats[2 * (n * NPART + p)] - Mg);
    Zg += w[p] * pstats[2 * (n * NPART + p) + 1];
  }
  float invZ = 1.f / Zg;
  float4 acc = make_float4(0.f, 0.f, 0.f, 0.f);
#pragma unroll
  for (int p = 0; p < NPART; ++p) {
    float4 v = *(const float4*)(pC + ((long)(n * NPART + p)) * 1024 + t * 4);
    acc.x += w[p] * v.x; acc.y += w[p] * v.y;
    acc.z += w[p] * v.z; acc.w += w[p] * v.w;
  }
  acc.x *= invZ; acc.y *= invZ; acc.z *= invZ; acc.w *= invZ;
  *(float4*)(ctx + (long)n * 1024 + t * 4) = acc;
}

// inp0[n] = [ctx | emb | hidden0];  inp1[n, 512:1024] = hidden1
__global__ void k_build(const float* __restrict__ ctx, const int* __restrict__ x,
                        const float* __restrict__ emb, const float* __restrict__ hidden,
                        float* __restrict__ inp0, float* __restrict__ inp1) {
  int n = blockIdx.x, t = threadIdx.x;
  const float* er = emb + x[n] * EMB;
  const float* h0 = hidden + n * HD;
  const float* h1 = hidden + NB * HD + n * HD;
  for (int c = t; c < 1792; c += 256) {
    float v;
    if (c < 1024)      v = ctx[n * 1024 + c];
    else if (c < 1280) v = er[c - 1024];
    else               v = h0[c - 1280];
    inp0[n * 1792 + c] = v;
  }
  for (int c = 512 + t; c < 1024; c += 256) inp1[n * 1024 + c] = h1[c - 512];
}

// gates[32 x 2048] = A[32 x K] * [W1 | W2]^T + bih + bhh  via V_WMMA_F32_16X16X4_F32
// One wave per 16x16 output tile.  A-tile lane layout (ISA 7.12.2, 32-bit A 16x4):
//   lane<16: {K=k, K=k+1}; lane>=16: {K=k+2, K=k+3}; M = lane&15.
// D layout: VGPR r -> row (r + 8*(lane>>4)), col (lane&15).
__global__ void k_gates_wmma(const float* __restrict__ A, int K, int Ksplit,
                             const float* __restrict__ W1, const float* __restrict__ W2,
                             const float* __restrict__ bih, const float* __restrict__ bhh,
                             float* __restrict__ gates) {
  int lane = threadIdx.x;
  int gTile = blockIdx.x;   // 0..127
  int mTile = blockIdx.y;   // 0..1
  int hi = lane >> 4, lo = lane & 15;
  int koff = hi * 2;
  int m = mTile * 16 + lo;
  int g = gTile * 16 + lo;
  const float* arow  = A  + (long)m * K;
  const float* w1row = W1 + (long)g * Ksplit;
  const float* w2row = W2 + (long)g * (K - Ksplit);
  v8f acc = {};
  for (int k = 0; k < K; k += 4) {
    int ke = k + koff;
    v2f a = *(const v2f*)(arow + ke);
    v2f b = (ke < Ksplit) ? *(const v2f*)(w1row + ke)
                          : *(const v2f*)(w2row + (ke - Ksplit));
    acc = __builtin_amdgcn_wmma_f32_16x16x4_f32(
        /*neg_a=*/false, a, /*neg_b=*/false, b,
        /*c_mod=*/(short)0, acc, /*reuse_a=*/false, /*reuse_b=*/false);
  }
  float bias = bih[g] + bhh[g];
#pragma unroll
  for (int r = 0; r < 8; ++r) {
    int row = mTile * 16 + r + 8 * hi;
    gates[(long)row * G4 + g] = acc[r] + bias;
  }
}

// LSTM elementwise: i,f,g,o gate order (PyTorch)
__global__ void k_lstm_act(const float* __restrict__ gates, const float* __restrict__ c_prev,
                           float* __restrict__ h_out, float* __restrict__ c_out,
                           float* __restrict__ h_extra) {
  int idx = blockIdx.x * 256 + threadIdx.x;  // 0..16383 = n*512+h
  int n = idx >> 9, h = idx & 511;
  const float* gr = gates + (long)n * G4;
  float gi = gr[h], gf = gr[512 + h], gg = gr[1024 + h], go = gr[1536 + h];
  float si = 1.f / (1.f + expf(-gi));
  float sf = 1.f / (1.f + expf(-gf));
  float so = 1.f / (1.f + expf(-go));
  float cn = sf * c_prev[idx] + si * tanhf(gg);
  float hn = so * tanhf(cn);
  c_out[idx] = cn;
  h_out[idx] = hn;
  if (h_extra) h_extra[(long)n * 1024 + h] = hn;  // inp1 cols [0,512)
}

// pred[n][cls] = dot(h1[n], Wc[cls]) + bc[cls]
__global__ void k_pred(const float* __restrict__ h1, const float* __restrict__ Wc,
                       const float* __restrict__ bc, float* __restrict__ pred) {
  int b = blockIdx.x, lane = threadIdx.x;
  int n = b / 10, cls = b - n * 10;
  const float* hr = h1 + n * HD;
  const float* wr = Wc + cls * HD;
  float acc = 0.f;
  for (int j = lane; j < HD; j += 32) acc += hr[j] * wr[j];
  acc = warp_red_sum(acc);
  if (lane == 0) pred[n * 10 + cls] = acc + bc[cls];
}

extern "C" void kernel_launch(void* const* d_in, const int* in_sizes, int n_in,
                              void* d_out, int out_size, void* d_ws, size_t ws_size,
                              hipStream_t stream) {
  const int*   x      = (const int*)  d_in[0];
  const float* enc    = (const float*)d_in[1];
  const float* hidden = (const float*)d_in[2];
  const float* cell   = (const float*)d_in[3];
  const float* Wen    = (const float*)d_in[4];
  const float* be     = (const float*)d_in[5];
  const float* emb    = (const float*)d_in[6];
  const float* wih0   = (const float*)d_in[7];
  const float* whh0   = (const float*)d_in[8];
  const float* bih0   = (const float*)d_in[9];
  const float* bhh0   = (const float*)d_in[10];
  const float* wih1   = (const float*)d_in[11];
  const float* whh1   = (const float*)d_in[12];
  const float* bih1   = (const float*)d_in[13];
  const float* bhh1   = (const float*)d_in[14];
  const float* Wc     = (const float*)d_in[15];
  const float* bc     = (const float*)d_in[16];

  float* ws     = (float*)d_ws;
  float* hc     = ws;                         // 32
  float* pstats = hc + 32;                    // 512 (M,Z per partition)
  float* pC     = pstats + 2 * NB * NPART;    // 32*8*1024 = 262144
  float* ctx    = pC + NB * NPART * 1024;     // 32768
  float* inp0   = ctx + NB * 1024;            // 57344
  float* inp1   = inp0 + NB * 1792;           // 32768
  float* gates  = inp1 + NB * 1024;           // 65536

  float* out   = (float*)d_out;
  float* pred  = out;                // 320
  float* h0o   = out + 320;
  float* h1o   = h0o + NB * HD;
  float* c0o   = h1o + NB * HD;
  float* c1o   = c0o + NB * HD;

  k_hconst      <<<NB, 32, 0, stream>>>(hidden, Wen, hc);
  k_attn_fused  <<<NB * NPART, 256, 0, stream>>>(enc, Wen, hc, be, pC, pstats);
  k_attn_combine<<<NB, 256, 0, stream>>>(pC, pstats, ctx);
  k_build       <<<NB, 256, 0, stream>>>(ctx, x, emb, hidden, inp0, inp1);

  k_gates_wmma<<<dim3(128, 2), 32, 0, stream>>>(inp0, 1792, 1280, wih0, whh0, bih0, bhh0, gates);
  k_lstm_act  <<<64, 256, 0, stream>>>(gates, cell, h0o, c0o, inp1);
  k_gates_wmma<<<dim3(128, 2), 32, 0, stream>>>(inp1, 1024, 512, wih1, whh1, bih1, bhh1, gates);
  k_lstm_act  <<<64, 256, 0, stream>>>(gates, cell + NB * HD, h1o, c1o, nullptr);

  k_pred<<<NB * 10, 32, 0, stream>>>(h1o, Wc, bc, pred);
}